// MeanPool_26560077758830
// MI455X (gfx1250) — compile-verified
//
#include <hip/hip_runtime.h>

// Problem sizes (fixed by the reference)
#define B_ 8
#define N_ 128
#define D_ 256

// colsum work decomposition: 32 chunks per batch, 4 n-rows per chunk
#define NCHUNK 32
#define NPC 4

typedef __attribute__((ext_vector_type(2))) float v2f;
typedef __attribute__((ext_vector_type(8))) float v8f;

// ---------------------------------------------------------------------------
// Kernel 1: lam[b,n] = (s[b,0,n] + s[b,1,n] + onehot_mask) / (sum(s[b,0:2,:]) + 1e-6)
// grid = B_, block = N_ (128 threads = 4 waves, wave32)
// ---------------------------------------------------------------------------
__global__ void lam_kernel(const float* __restrict__ score,   // (B,3,N)
                           const int*   __restrict__ sub_graph,
                           float*       __restrict__ lam) {   // (B,N) ws
    int b = blockIdx.x;
    int n = threadIdx.x;
    float s0 = score[(b * 3 + 0) * N_ + n];
    float s1 = score[(b * 3 + 1) * N_ + n];
    float base = s0 + s1;

    // wave32 reduction then cross-wave via LDS
    float v = base;
    #pragma unroll
    for (int off = 16; off > 0; off >>= 1)
        v += __shfl_xor(v, off, 32);

    __shared__ float sw[4];
    int lane = n & 31, wid = n >> 5;
    if (lane == 0) sw[wid] = v;
    __syncthreads();
    float tot = sw[0] + sw[1] + sw[2] + sw[3] + 1e-6f;

    int sg = sub_graph[b];
    float mask = (n == sg && n != 0) ? 1.0f : 0.0f;
    lam[b * N_ + n] = (base + mask) / tot;
}

// ---------------------------------------------------------------------------
// Kernel 2: lam-weighted column sums of s_e (the 134 MB stream).
// grid = B_*NCHUNK, block = 256.  Thread = (m-group 0..3, d-quad 0..63).
// Each thread: float4 loads, register accumulate; LDS combine across m-groups.
// partial[(b*NCHUNK+chunk)*D + d] = sum over this chunk's 4 n-rows of
//   lam[b,n] * sum_m s_e[b,n,m,d]
// ---------------------------------------------------------------------------
__global__ __launch_bounds__(256)
void colsum_kernel(const float* __restrict__ s_e,      // (B,N,N,D)
                   const float* __restrict__ lam,      // (B,N)
                   float*       __restrict__ partial) {
    int blk   = blockIdx.x;
    int b     = blk / NCHUNK;
    int chunk = blk % NCHUNK;
    int n0    = chunk * NPC;

    int tid = threadIdx.x;
    int mg  = tid >> 6;        // 0..3  -> m range [mg*32, mg*32+32)
    int dq  = tid & 63;        // 0..63 -> d quad
    int d   = dq * 4;

    float ax = 0.f, ay = 0.f, az = 0.f, aw = 0.f;
    for (int i = 0; i < NPC; ++i) {
        int n = n0 + i;
        float w = lam[b * N_ + n];
        const float4* p = (const float4*)(s_e +
            ((((size_t)b * N_ + n) * N_ + (size_t)mg * 32) * D_ + d));
        float rx = 0.f, ry = 0.f, rz = 0.f, rw = 0.f;
        #pragma unroll 8
        for (int m = 0; m < 32; ++m) {
            float4 v = p[(size_t)m * (D_ / 4)];
            rx += v.x; ry += v.y; rz += v.z; rw += v.w;
        }
        ax += w * rx; ay += w * ry; az += w * rz; aw += w * rw;
    }

    __shared__ float4 red[256];
    red[tid] = make_float4(ax, ay, az, aw);
    __syncthreads();
    if (tid < 64) {
        float4 a0 = red[tid];
        float4 a1 = red[tid + 64];
        float4 a2 = red[tid + 128];
        float4 a3 = red[tid + 192];
        float4 s = make_float4(a0.x + a1.x + a2.x + a3.x,
                               a0.y + a1.y + a2.y + a3.y,
                               a0.z + a1.z + a2.z + a3.z,
                               a0.w + a1.w + a2.w + a3.w);
        *(float4*)(partial + ((size_t)blk * D_ + d)) = s;
    }
}

// ---------------------------------------------------------------------------
// Kernel 3: finalize.
//   sv_g[b,d] = sum_n lam[b,n] * s_v[b,n,d]   via V_WMMA_F32_16X16X4_F32
//   out[b,d]  = (N * sv_g + sum_chunks partial) / num_node[b]^2
// grid = B_, block = 512 (16 waves; wave w owns d-tile [w*16, w*16+16)).
// A = all-ones 16x4, B[k,j] = lam[n0+k] * s_v[b, n0+k, dbase+j]; C accumulates
// over 32 K-steps. With all-ones A every row of D equals the weighted column
// sum, independent of the hardware's K-slot ordering, and the documented C/D
// layout gives lane L the value for column (L & 15).
// ---------------------------------------------------------------------------
__global__ __launch_bounds__(512)
void finalize_kernel(const float* __restrict__ s_v,      // (B,N,D)
                     const float* __restrict__ lam,      // (B,N)
                     const float* __restrict__ partial,  // (B*NCHUNK, D)
                     const float* __restrict__ num_node, // (B,)
                     float*       __restrict__ out) {    // (B,1,D)
    int b    = blockIdx.x;
    int tid  = threadIdx.x;
    int wave = tid >> 5;           // 0..15
    int lane = tid & 31;
    int col  = lane & 15;
    int half = lane >> 4;
    int dbase = wave * 16;

    const float* lamb = lam + b * N_;
    const float* svb  = s_v + (size_t)b * N_ * D_;

    v8f c = {};
    v2f a = {1.0f, 1.0f};

    for (int n0 = 0; n0 < N_; n0 += 4) {
        int na = n0 + half;        // rows for B-operand VGPR0
        int nb = na + 2;           // rows for B-operand VGPR1
        v2f bm;
        bm[0] = lamb[na] * svb[(size_t)na * D_ + dbase + col];
        bm[1] = lamb[nb] * svb[(size_t)nb * D_ + dbase + col];
        // D = A(ones) x B + C  -> every D row = weighted column sum, C chains
        c = __builtin_amdgcn_wmma_f32_16x16x4_f32(
                /*neg_a=*/false, a, /*neg_b=*/false, bm,
                /*c_mod=*/(short)0, c, /*reuse_a=*/false, /*reuse_b=*/false);
    }

    float sv_g = c[0];             // lane L: row (L<16 ? 0 : 8), col L&15 — all rows equal

    if (lane < 16) {
        int d = dbase + col;
        float se = 0.f;
        #pragma unroll 8
        for (int k = 0; k < NCHUNK; ++k)
            se += partial[(size_t)(b * NCHUNK + k) * D_ + d];
        float nn = num_node[b];
        out[b * D_ + d] = ((float)N_ * sv_g + se) / (nn * nn);
    }
}

// ---------------------------------------------------------------------------
extern "C" void kernel_launch(void* const* d_in, const int* in_sizes, int n_in,
                              void* d_out, int out_size, void* d_ws, size_t ws_size,
                              hipStream_t stream) {
    const float* s_v       = (const float*)d_in[0];  // (B,N,D)
    const float* s_e       = (const float*)d_in[1];  // (B,N,N,D)
    const int*   sub_graph = (const int*)d_in[2];    // (B,)
    const float* s_e_score = (const float*)d_in[3];  // (B,3,N)
    const float* num_node  = (const float*)d_in[4];  // (B,) float
    float* out = (float*)d_out;                      // (B,1,D) flat

    float* lam     = (float*)d_ws;                   // B*N floats
    float* partial = lam + B_ * N_;                  // B*NCHUNK*D floats (256 KB)

    lam_kernel<<<B_, N_, 0, stream>>>(s_e_score, sub_graph, lam);
    colsum_kernel<<<B_ * NCHUNK, 256, 0, stream>>>(s_e, lam, partial);
    finalize_kernel<<<B_, 512, 0, stream>>>(s_v, lam, partial, num_node, out);
}